// GCNClassifier_64209761075709
// MI455X (gfx1250) — compile-verified
//
#include <hip/hip_runtime.h>

#define N_NODES   100000
#define N_EDGES   1600000
#define IN_DIM    128
#define HIDDEN    128
#define N_CLASSES 16

typedef __attribute__((ext_vector_type(2))) float v2f;
typedef __attribute__((ext_vector_type(8))) float v8f;

// ---------------- init: degrees <- 1 (self loop), agg <- 0, hg_sum <- 0 -------
__global__ void k_init(float* __restrict__ agg, float* __restrict__ deg_out,
                       float* __restrict__ deg_in, float* __restrict__ hg_sum) {
  size_t tid = (size_t)blockIdx.x * blockDim.x + threadIdx.x;
  if (tid < (size_t)N_NODES * HIDDEN) agg[tid] = 0.0f;
  if (tid < N_NODES) { deg_out[tid] = 1.0f; deg_in[tid] = 1.0f; }
  if (tid < HIDDEN)  hg_sum[tid] = 0.0f;
}

// ---------------- per-edge degree accumulation --------------------------------
__global__ void k_degree(const int* __restrict__ src, const int* __restrict__ dst,
                         float* __restrict__ deg_out, float* __restrict__ deg_in) {
  int e = blockIdx.x * blockDim.x + threadIdx.x;
  if (e >= N_EDGES) return;
  atomicAdd(&deg_out[src[e]], 1.0f);
  atomicAdd(&deg_in[dst[e]],  1.0f);
}

// ---------------- deg -> deg^-1/2 (deg >= 1 due to self loops) ----------------
__global__ void k_norm(float* __restrict__ deg_out, float* __restrict__ deg_in) {
  int i = blockIdx.x * blockDim.x + threadIdx.x;
  if (i >= N_NODES) return;
  deg_out[i] = 1.0f / sqrtf(deg_out[i]);
  deg_in[i]  = 1.0f / sqrtf(deg_in[i]);
}

// ---------------- SpMM: one wave32 per edge, float4 per lane ------------------
// agg[dst] += h[src] * norm_out[src].  h (51.2MB) is L2-resident on MI455X.
__global__ void k_spmm(const float* __restrict__ h, const int* __restrict__ src,
                       const int* __restrict__ dst, const float* __restrict__ norm_out,
                       float* __restrict__ agg) {
  size_t gid = (size_t)blockIdx.x * blockDim.x + threadIdx.x;
  int e    = (int)(gid >> 5);
  int lane = (int)(gid & 31);
  if (e >= N_EDGES) return;
  int s = src[e], d = dst[e];
  float w = norm_out[s];
  float4 v = ((const float4*)(h + (size_t)s * IN_DIM))[lane];
  float* o = agg + (size_t)d * IN_DIM + lane * 4;
  atomicAdd(o + 0, v.x * w);
  atomicAdd(o + 1, v.y * w);
  atomicAdd(o + 2, v.z * w);
  atomicAdd(o + 3, v.w * w);
}

// ---------------- self-loop add + in-degree scale -----------------------------
// agg[i] = (agg[i] + h[i]*norm_out[i]) * norm_in[i]
__global__ void k_selfloop_scale(const float* __restrict__ h,
                                 const float* __restrict__ norm_out,
                                 const float* __restrict__ norm_in,
                                 float* __restrict__ agg) {
  size_t gid = (size_t)blockIdx.x * blockDim.x + threadIdx.x;
  int node = (int)(gid >> 5);
  int lane = (int)(gid & 31);
  if (node >= N_NODES) return;
  float wo = norm_out[node], wi = norm_in[node];
  float4 hv = ((const float4*)(h + (size_t)node * IN_DIM))[lane];
  float4* ap = ((float4*)(agg + (size_t)node * IN_DIM)) + lane;
  float4 a = *ap;
  a.x = (a.x + hv.x * wo) * wi;
  a.y = (a.y + hv.y * wo) * wi;
  a.z = (a.z + hv.z * wo) * wi;
  a.w = (a.w + hv.w * wo) * wi;
  *ap = a;
}

// ---------------- fused GEMM(relu(agg@W1+b1)) + column-sum via V_WMMA_F32_16X16X4_F32
// Block = 256 threads = 8 waves. Block handles a 16-node tile; wave w handles
// the 16-column tile [16w, 16w+16).  K=128 in 32 WMMA steps of K=4.
// A (16x4 f32): lanes 0-15 carry K={k,k+1}, lanes 16-31 carry K={k+2,k+3}, M=lane%16.
// B (4x16 f32): mirrored K assignment across lane halves, N=lane%16.
// C/D (16x16 f32, v8f): element r, lane L -> row M = r + 8*(L>=16), col = L%16.
// Full tiles (the only case when 16|N_NODES) take a branch-free unmasked path;
// a masked path (clamped row + multiply-by-0, EXEC stays all-ones) guards the
// general tail case.  The tile-full test is block-uniform -> no divergence.
__global__ __launch_bounds__(256) void k_gemm_mean(const float* __restrict__ agg,
                                                   const float* __restrict__ W1,
                                                   const float* __restrict__ b1,
                                                   float* __restrict__ hg_sum) {
  const int lane = threadIdx.x & 31;
  const int wave = threadIdx.x >> 5;     // 0..7 -> column tile
  const int col0 = wave * 16;
  const int row0 = blockIdx.x * 16;      // node tile
  const int half = lane >> 4;            // 0: K+{0,1}, 1: K+{2,3}
  const int l16  = lane & 15;

  const int node   = row0 + l16;         // A row owned by this lane (M = lane%16)
  const float* bcol = W1 + col0 + l16 + (size_t)(half * 2) * HIDDEN;
  const float bias  = b1[col0 + l16];

  v8f c = {0.f, 0.f, 0.f, 0.f, 0.f, 0.f, 0.f, 0.f};
  float colsum = 0.0f;

  if (row0 + 16 <= N_NODES) {            // uniform: full 16-node tile (hot path)
    const float* arow = agg + (size_t)node * IN_DIM + half * 2;  // 8B aligned
    #pragma unroll 8
    for (int k = 0; k < IN_DIM; k += 4) {
      v2f a = *(const v2f*)(arow + k);   // A[M][ka], A[M][ka+1] (b64 load)
      v2f b;
      b.x = bcol[(size_t)k * HIDDEN];            // W1[ka  ][col0+l16]
      b.y = bcol[(size_t)k * HIDDEN + HIDDEN];   // W1[ka+1][col0+l16]
      // 8 args: (neg_a, A, neg_b, B, c_mod, C, reuse_a, reuse_b)
      c = __builtin_amdgcn_wmma_f32_16x16x4_f32(false, a, false, b, (short)0, c,
                                                false, false);
    }
    #pragma unroll
    for (int r = 0; r < 8; ++r) {
      float v = c[r] + bias;
      colsum += (v > 0.0f ? v : 0.0f);
    }
  } else {                               // tail tile: mask rows branch-free
    const float validf = (node < N_NODES) ? 1.0f : 0.0f;
    const int   nodeC  = (node < N_NODES) ? node : 0;
    const float* arow  = agg + (size_t)nodeC * IN_DIM + half * 2;
    #pragma unroll 8
    for (int k = 0; k < IN_DIM; k += 4) {
      v2f a = *(const v2f*)(arow + k);
      a = a * validf;
      v2f b;
      b.x = bcol[(size_t)k * HIDDEN];
      b.y = bcol[(size_t)k * HIDDEN + HIDDEN];
      c = __builtin_amdgcn_wmma_f32_16x16x4_f32(false, a, false, b, (short)0, c,
                                                false, false);
    }
    #pragma unroll
    for (int r = 0; r < 8; ++r) {
      const int nd = row0 + r + half * 8;
      float v = c[r] + bias;
      v = v > 0.0f ? v : 0.0f;
      if (nd < N_NODES) colsum += v;
    }
  }

  // lanes L and L+16 hold the same column (rows 0-7 vs 8-15)
  colsum += __shfl_down(colsum, 16, 32);
  if (half == 0) atomicAdd(&hg_sum[col0 + l16], colsum);
}

// ---------------- heads: out[0:16]=hg@Wc1+bc1, out[16:32]=hg@Wc2+bc2 ----------
__global__ void k_head(const float* __restrict__ hg_sum,
                       const float* __restrict__ Wc1, const float* __restrict__ bc1,
                       const float* __restrict__ Wc2, const float* __restrict__ bc2,
                       float* __restrict__ out) {
  const int t = threadIdx.x;             // 0..31
  if (t >= 2 * N_CLASSES) return;
  const float* Wc = (t < N_CLASSES) ? Wc1 : Wc2;
  const float* bc = (t < N_CLASSES) ? bc1 : bc2;
  const int j = t & (N_CLASSES - 1);
  const float invN = 1.0f / (float)N_NODES;
  float acc = 0.0f;
  for (int cidx = 0; cidx < HIDDEN; ++cidx)
    acc += (hg_sum[cidx] * invN) * Wc[cidx * N_CLASSES + j];
  out[t] = acc + bc[j];
}

extern "C" void kernel_launch(void* const* d_in, const int* in_sizes, int n_in,
                              void* d_out, int out_size, void* d_ws, size_t ws_size,
                              hipStream_t stream) {
  const float* h   = (const float*)d_in[0];
  const int*   src = (const int*)  d_in[1];
  const int*   dst = (const int*)  d_in[2];
  const float* W1  = (const float*)d_in[3];
  const float* b1  = (const float*)d_in[4];
  const float* Wc1 = (const float*)d_in[5];
  const float* bc1 = (const float*)d_in[6];
  const float* Wc2 = (const float*)d_in[7];
  const float* bc2 = (const float*)d_in[8];
  float* out = (float*)d_out;

  // workspace layout (floats): agg[N*128] | deg_out[N] | deg_in[N] | hg_sum[128]
  float* agg     = (float*)d_ws;
  float* deg_out = agg + (size_t)N_NODES * HIDDEN;
  float* deg_in  = deg_out + N_NODES;
  float* hg_sum  = deg_in + N_NODES;

  const int B = 256;
  const size_t aggElems = (size_t)N_NODES * HIDDEN;            // 12.8M
  k_init<<<(unsigned)((aggElems + B - 1) / B), B, 0, stream>>>(agg, deg_out, deg_in, hg_sum);
  k_degree<<<(N_EDGES + B - 1) / B, B, 0, stream>>>(src, dst, deg_out, deg_in);
  k_norm<<<(N_NODES + B - 1) / B, B, 0, stream>>>(deg_out, deg_in);

  const size_t spmmThreads = (size_t)N_EDGES * 32;             // one wave per edge
  k_spmm<<<(unsigned)((spmmThreads + B - 1) / B), B, 0, stream>>>(h, src, dst, deg_out, agg);

  const size_t scaleThreads = (size_t)N_NODES * 32;
  k_selfloop_scale<<<(unsigned)((scaleThreads + B - 1) / B), B, 0, stream>>>(h, deg_out, deg_in, agg);

  k_gemm_mean<<<(N_NODES + 15) / 16, 256, 0, stream>>>(agg, W1, b1, hg_sum);

  k_head<<<1, 32, 0, stream>>>(hg_sum, Wc1, bc1, Wc2, bc2, out);
}